// SetTransformer4_83494164234888
// MI455X (gfx1250) — compile-verified
//
#include <hip/hip_runtime.h>
#include <cmath>

#define DIMD 128
#define NHEAD 4
#define BATCH 16
#define SEQ 1000
#define NIND 32

typedef float v2f __attribute__((ext_vector_type(2)));
typedef float v8f __attribute__((ext_vector_type(8)));

#if defined(__gfx1250__) && __has_builtin(__builtin_amdgcn_global_load_async_to_lds_b32) && \
    __has_builtin(__builtin_amdgcn_s_wait_asynccnt)
#define USE_ASYNC 1
#else
#define USE_ASYNC 0
#endif

#if USE_ASYNC
#define AS_GLOBAL __attribute__((address_space(1)))
#define AS_LDS __attribute__((address_space(3)))
__device__ __forceinline__ AS_LDS int* to_lds(void* p) {
  return (AS_LDS int*)(uintptr_t)p;   // generic LDS addr[31:0] is the LDS offset
}
__device__ __forceinline__ AS_GLOBAL int* to_glob(const void* p) {
  return (AS_GLOBAL int*)(uintptr_t)p;
}
#endif

// ---------------- in_proj: y[r,c] = x[r,0]*W[0,c] + x[r,1]*W[1,c] + b[c] ----
__global__ void in_proj_kernel(const float* __restrict__ x,
                               const float* __restrict__ Wb,
                               float* __restrict__ y, int rows) {
  int idx = blockIdx.x * blockDim.x + threadIdx.x;
  if (idx >= rows * DIMD) return;
  int r = idx >> 7, c = idx & 127;
  y[idx] = x[2 * r] * Wb[c] + x[2 * r + 1] * Wb[DIMD + c] + Wb[2 * DIMD + c];
}

// ---------------- 128x128 linear via fp32 WMMA -------------------------------
// Y[M,128] = post( X[M,128] @ W[128,128] + b ),  W row-major at Wp, bias at Wp+128*128
// mode bit0: zero rows where (row % SEQ) >= lens[row / SEQ]   (mask_rows; only big acts)
// mode bit1: Y = Res + relu(X@W + b)                          (o-projection)
__global__ void lin128_wmma(const float* __restrict__ X,
                            const float* __restrict__ Wp,
                            const float* __restrict__ Res,
                            float* __restrict__ Y,
                            const int* __restrict__ lens,
                            int M, int mode) {
  int wave = (blockIdx.x * blockDim.x + threadIdx.x) >> 5;
  int nTileM = M >> 4;
  if (wave >= nTileM * 8) return;            // wave-uniform exit (EXEC stays all-1s)
  int tm = wave >> 3, tn = wave & 7;
  int lane = threadIdx.x & 31;
  int laneLo = lane & 15;
  int half = lane >> 4;
  int r0 = tm << 4, c0 = tn << 4;
  const float* bias = Wp + DIMD * DIMD;
  const float* xrow = X + (size_t)(r0 + laneLo) * DIMD;
  v8f acc = {};
#pragma unroll 4
  for (int k0 = 0; k0 < DIMD; k0 += 4) {
    int kA = k0 + half * 2;                  // lanes 0-15: K=k0,k0+1 ; lanes 16-31: K=k0+2,k0+3
    v2f a, b;
    a.x = xrow[kA];
    a.y = xrow[kA + 1];
    b.x = Wp[(size_t)kA * DIMD + c0 + laneLo];
    b.y = Wp[(size_t)(kA + 1) * DIMD + c0 + laneLo];
    acc = __builtin_amdgcn_wmma_f32_16x16x4_f32(false, a, false, b, (short)0,
                                                acc, false, false);
  }
  int col = c0 + laneLo;
  float bcol = bias[col];
#pragma unroll
  for (int r = 0; r < 8; ++r) {
    int row = r0 + r + half * 8;             // C/D layout: VGPR r -> M=r (lo half), M=r+8 (hi half)
    float v = acc[r] + bcol;
    if (mode & 1) {
      int bb = row / SEQ;                    // constant divisor -> mul/shift
      int n = row - bb * SEQ;
      if (n >= lens[bb]) v = 0.f;
    }
    if (mode & 2) {
      v = Res[(size_t)row * DIMD + col] + fmaxf(v, 0.f);
    }
    Y[(size_t)row * DIMD + col] = v;
  }
}

// ---- stage one 16-key x 32-channel K and V head tile into LDS (stride 34) ----
__device__ __forceinline__ void stage_tile(const float* __restrict__ gK,
                                           const float* __restrict__ gV,
                                           float* ldsK, float* ldsV,
                                           int kbase, int Lk, int lane) {
#pragma unroll
  for (int j = 0; j < 16; ++j) {
    int key = kbase + j;
    if (key >= Lk) key = Lk - 1;             // clamp; invalid keys neutralized later
#if USE_ASYNC
    __builtin_amdgcn_global_load_async_to_lds_b32(
        to_glob(gK + (size_t)key * DIMD + lane), to_lds(ldsK + j * 34 + lane), 0, 0);
    __builtin_amdgcn_global_load_async_to_lds_b32(
        to_glob(gV + (size_t)key * DIMD + lane), to_lds(ldsV + j * 34 + lane), 0, 0);
#else
    ldsK[j * 34 + lane] = gK[(size_t)key * DIMD + lane];
    ldsV[j * 34 + lane] = gV[(size_t)key * DIMD + lane];
#endif
  }
}

__device__ __forceinline__ void wait_tiles() {
#if USE_ASYNC
  __builtin_amdgcn_s_wait_asynccnt(0);
#endif
  __syncthreads();
}

// ---------------- WMMA flash-attention MAB core ------------------------------
// One wave (32 threads, own workgroup) per (batch b, 16-query tile qt, head h).
// O[b,q, h*32 + 0..31] = Resid[...] + softmax(Qh_h Kh_h^T / sqrt(128)) @ Vh_h
// flags bit0: Qh batch-broadcast (indexed [q,:]) ; bit1: Resid batch-broadcast
__global__ void attn_wmma(const float* __restrict__ Qh,
                          const float* __restrict__ Kh,
                          const float* __restrict__ Vh,
                          const float* __restrict__ Resid,
                          float* __restrict__ O,
                          int Lq, int Lk, int flags) {
  __shared__ float Kt[2][16 * 34];  // ping-pong K tile (async-staged)
  __shared__ float Vt[2][16 * 34];  // ping-pong V tile
  __shared__ float Slds[16 * 17];   // S / P tile
  __shared__ float rs[16];          // per-row online rescale factor
  __shared__ float rowl[16];        // per-row 1/sum
  int nQT = (Lq + 15) >> 4;
  int wid = blockIdx.x;
  int h = wid & 3;
  int t = wid >> 2;
  int qt = t % nQT;
  int b = t / nQT;
  int lane = threadIdx.x;
  int laneLo = lane & 15;
  int half = lane >> 4;
  const float scale = 0.08838834764831843f;  // 1/sqrt(dim_V) = 1/sqrt(128)

  // Preload Q A-fragments for this head: 8 K-steps of 4 channels each.
  // A layout (16x4 f32): lanes 0-15 = rows, halves carry K pairs {0,1}/{2,3}.
  int qrow = qt * 16 + laneLo;
  int qr = qrow < Lq ? qrow : Lq - 1;
  size_t qbase = ((flags & 1) ? (size_t)qr : ((size_t)b * Lq + qr)) * DIMD + h * 32;
  v2f qa[8];
#pragma unroll
  for (int s = 0; s < 8; ++s) {
    int kA = s * 4 + half * 2;
    qa[s].x = Qh[qbase + kA];
    qa[s].y = Qh[qbase + kA + 1];
  }

  float m = -1e30f, l = 0.f;
  v8f acc0 = {}, acc1 = {};
  int nKT = (Lk + 15) >> 4;
  const float* KhB = Kh + (size_t)b * Lk * DIMD + h * 32;
  const float* VhB = Vh + (size_t)b * Lk * DIMD + h * 32;

  int bi = 0;
  stage_tile(KhB, VhB, &Kt[0][0], &Vt[0][0], 0, Lk, lane);

  for (int kt = 0; kt < nKT; ++kt) {
    int kbase = kt * 16;
    wait_tiles();  // current tile resident; prior iteration's LDS readers retired
    if (kt + 1 < nKT)  // overlap next tile's async copy with this tile's WMMAs
      stage_tile(KhB, VhB, &Kt[bi ^ 1][0], &Vt[bi ^ 1][0], kbase + 16, Lk, lane);

    // ---- S = Q K^T : 8 WMMAs over the 32 head channels (K from LDS) ----
    const float* kr = &Kt[bi][laneLo * 34];
    v8f st = {};
#pragma unroll
    for (int s = 0; s < 8; ++s) {
      int kA = s * 4 + half * 2;
      v2f bf;
      bf.x = kr[kA];
      bf.y = kr[kA + 1];
      st = __builtin_amdgcn_wmma_f32_16x16x4_f32(false, qa[s], false, bf, (short)0,
                                                 st, false, false);
    }
#pragma unroll
    for (int r = 0; r < 8; ++r)
      Slds[(r + half * 8) * 17 + laneLo] = st[r] * scale;
    __syncthreads();

    // ---- online softmax bookkeeping: lane owns row laneLo (halves mirror) ----
    int kvalid = Lk - kbase;
    if (kvalid > 16) kvalid = 16;
    float tmax = -1e30f;
    for (int n = 0; n < kvalid; ++n) tmax = fmaxf(tmax, Slds[laneLo * 17 + n]);
    float mnew = fmaxf(m, tmax);
    float f = expf(m - mnew);                  // first tile: exp(-1e30-x) -> 0
    float ssum = 0.f;
    if (half == 0) {
#pragma unroll
      for (int n = 0; n < 16; ++n) {
        float p = (n < kvalid) ? expf(Slds[laneLo * 17 + n] - mnew) : 0.f;
        Slds[laneLo * 17 + n] = p;             // S tile becomes P tile
        ssum += p;
      }
      rs[laneLo] = f;
    }
    __syncthreads();
    m = mnew;
    l = l * f + ssum;                          // meaningful on half==0 lanes

    // ---- rescale accumulators by per-row factor ----
#pragma unroll
    for (int r = 0; r < 8; ++r) {
      float fr = rs[r + half * 8];
      acc0[r] *= fr;
      acc1[r] *= fr;
    }
    // ---- acc += P(16x16) @ V(16x32): P reload performs D->A transpose ----
    const float* vt = &Vt[bi][0];
#pragma unroll
    for (int kk = 0; kk < 16; kk += 4) {
      v2f pa;
      pa.x = Slds[laneLo * 17 + kk + half * 2];
      pa.y = Slds[laneLo * 17 + kk + half * 2 + 1];
      int row0 = (kk + half * 2) * 34;         // key rows within tile (clamped at stage)
      v2f bv0, bv1;
      bv0.x = vt[row0 + laneLo];
      bv0.y = vt[row0 + 34 + laneLo];
      bv1.x = vt[row0 + 16 + laneLo];
      bv1.y = vt[row0 + 34 + 16 + laneLo];
      acc0 = __builtin_amdgcn_wmma_f32_16x16x4_f32(false, pa, false, bv0, (short)0,
                                                   acc0, false, false);
      acc1 = __builtin_amdgcn_wmma_f32_16x16x4_f32(false, pa, false, bv1, (short)0,
                                                   acc1, false, false);
    }
    bi ^= 1;
  }

  // ---- finalize: divide by row sums, add per-head residual, store ----
  if (half == 0) rowl[laneLo] = 1.f / l;
  __syncthreads();
#pragma unroll
  for (int r = 0; r < 8; ++r) {
    int qv = qt * 16 + r + half * 8;
    if (qv < Lq) {
      float li = rowl[r + half * 8];
      size_t oi = ((size_t)b * Lq + qv) * DIMD + h * 32 + laneLo;
      size_t ri = ((flags & 2) ? (size_t)qv : ((size_t)b * Lq + qv)) * DIMD + h * 32 + laneLo;
      O[oi]      = Resid[ri]      + acc0[r] * li;
      O[oi + 16] = Resid[ri + 16] + acc1[r] * li;
    }
  }
}

// ---------------- output head: out[b] = dot(Xp[b,:], Wout) + bout ------------
__global__ void out_kernel(const float* __restrict__ Xp,
                           const float* __restrict__ Wb,
                           float* __restrict__ out) {
  __shared__ float red[128];
  int b = blockIdx.x, tid = threadIdx.x;
  red[tid] = Xp[b * DIMD + tid] * Wb[tid];
  __syncthreads();
  for (int o = 64; o > 0; o >>= 1) {
    if (tid < o) red[tid] += red[tid + o];
    __syncthreads();
  }
  if (tid == 0) out[b] = red[0] + Wb[DIMD];
}

extern "C" void kernel_launch(void* const* d_in, const int* in_sizes, int n_in,
                              void* d_out, int out_size, void* d_ws, size_t ws_size,
                              hipStream_t stream) {
  (void)in_sizes; (void)n_in; (void)out_size; (void)ws_size;
  const float* x1 = (const float*)d_in[0];
  const float* x2 = (const float*)d_in[1];
  const float* P  = (const float*)d_in[2];   // params flattened in insertion order
  const int* lens = (const int*)d_in[3];
  float* out = (float*)d_out;

  constexpr int LIN = DIMD * DIMD + DIMD;            // 16512
  constexpr int MAB = 4 * LIN;                       // 66048  (q,k,v,o)
  constexpr int ISB = NIND * DIMD + 2 * MAB;         // 136192 (I, mab0, mab1)
  constexpr int OFF_TRANS = 384 + 2 * ISB;           // after in_proj + dead enc stack
  constexpr int OFF_E2 = OFF_TRANS + MAB;
  constexpr int OFF_PMA = OFF_E2 + 2 * ISB;          // S then mab
  constexpr int OFF_OUT = OFF_PMA + DIMD + MAB;      // W(128) then b(1)

  const size_t BIG = (size_t)BATCH * SEQ * DIMD;     // 2,048,000 floats
  float* A  = (float*)d_ws;      // x1p / X
  float* Bf = A + BIG;           // x2p / attn O / X
  float* C  = Bf + BIG;          // Qh
  float* Dq = C + BIG;           // Kh / attn O
  float* E  = Dq + BIG;          // Vh
  float* s0 = E + BIG;           // QhI   (32 x 128)
  float* s1 = s0 + NIND * DIMD;              // OI  (B x 32 x 128)
  float* s2 = s1 + BATCH * NIND * DIMD;      // Hm
  float* s3 = s2 + BATCH * NIND * DIMD;      // Kh2
  float* s4 = s3 + BATCH * NIND * DIMD;      // Vh2
  float* s5 = s4 + BATCH * NIND * DIMD;      // QhS (16 x 128, row 0 valid)
  float* s6 = s5 + 16 * DIMD;                // OS  (B x 128)
  float* s7 = s6 + 16 * DIMD;                // XP  (B x 128)

  auto lin = [&](const float* X, const float* Wp, const float* Res, float* Y,
                 int M, int mode) {
    int tiles = (M >> 4) * 8;
    lin128_wmma<<<(tiles + 3) / 4, 128, 0, stream>>>(X, Wp, Res, Y, lens, M, mode);
  };
  auto attn = [&](const float* Qh, const float* Kh, const float* Vh,
                  const float* Res, float* O, int Lq, int Lk, int flags) {
    int nQT = (Lq + 15) / 16;
    attn_wmma<<<BATCH * nQT * NHEAD, 32, 0, stream>>>(Qh, Kh, Vh, Res, O, Lq, Lk, flags);
  };

  const int Mbig = BATCH * SEQ;  // 16000 rows
  // 1. input projections
  in_proj_kernel<<<(Mbig * DIMD) / 256, 256, 0, stream>>>(x1, P, A, Mbig);
  in_proj_kernel<<<(Mbig * DIMD) / 256, 256, 0, stream>>>(x2, P, Bf, Mbig);

  // 2. transition MAB: Q=x1p, K=x2p, mask Q and K
  lin(A,  P + OFF_TRANS + 0 * LIN, nullptr, C,  Mbig, 1);  // Qh (masked)
  lin(Bf, P + OFF_TRANS + 1 * LIN, nullptr, Dq, Mbig, 1);  // Kh (masked)
  lin(Bf, P + OFF_TRANS + 2 * LIN, nullptr, E,  Mbig, 1);  // Vh (masked)
  attn(C, Dq, E, A, Bf, SEQ, SEQ, 0);                      // O (resid = raw x1p)
  lin(Bf, P + OFF_TRANS + 3 * LIN, Bf, A, Mbig, 2);        // X = O + relu(O Wo + b)

  float* Xc = A;
  float* Xn = Bf;
  // 3. two ISABs (enc2)
  for (int i = 0; i < 2; ++i) {
    int iso = OFF_E2 + i * ISB;
    int m0 = iso + NIND * DIMD;
    int m1 = m0 + MAB;
    // mab0: Q = I (batch-broadcast), K = X, mask K
    lin(P + iso, P + m0 + 0 * LIN, nullptr, s0, NIND, 0);      // QhI (once)
    lin(Xc, P + m0 + 1 * LIN, nullptr, Dq, Mbig, 1);           // Kh
    lin(Xc, P + m0 + 2 * LIN, nullptr, E,  Mbig, 1);           // Vh
    attn(s0, Dq, E, P + iso, s1, NIND, SEQ, 3);                // OI (Q,resid broadcast)
    lin(s1, P + m0 + 3 * LIN, s1, s2, BATCH * NIND, 2);        // Hm
    // mab1: Q = X (masked), K = Hm
    lin(Xc, P + m1 + 0 * LIN, nullptr, C, Mbig, 1);            // Qh
    lin(s2, P + m1 + 1 * LIN, nullptr, s3, BATCH * NIND, 0);   // Kh2
    lin(s2, P + m1 + 2 * LIN, nullptr, s4, BATCH * NIND, 0);   // Vh2
    attn(C, s3, s4, Xc, Dq, SEQ, NIND, 0);                     // O (resid = raw X)
    lin(Dq, P + m1 + 3 * LIN, Dq, Xn, Mbig, 2);                // X
    float* t = Xc; Xc = Xn; Xn = t;
  }

  // 4. PMA: Q = S (1 row, batch-broadcast), K = X, mask K
  lin(P + OFF_PMA, P + OFF_PMA + DIMD + 0 * LIN, nullptr, s5, 16, 0);  // QhS
  lin(Xc, P + OFF_PMA + DIMD + 1 * LIN, nullptr, Dq, Mbig, 1);
  lin(Xc, P + OFF_PMA + DIMD + 2 * LIN, nullptr, E,  Mbig, 1);
  attn(s5, Dq, E, P + OFF_PMA, s6, 1, SEQ, 3);
  lin(s6, P + OFF_PMA + DIMD + 3 * LIN, s6, s7, 16, 2);

  // 5. output head
  out_kernel<<<BATCH, 128, 0, stream>>>(s7, P + OFF_OUT, out);
}